// GCN_24550033064030
// MI455X (gfx1250) — compile-verified
//
#include <hip/hip_runtime.h>

typedef __attribute__((ext_vector_type(2))) float v2f;
typedef __attribute__((ext_vector_type(8))) float v8f;

#define N_NODES 100000
#define N_EDGES 3200000
#define IN_DIM  128
#define HD_DIM  64
#define OUT_DIM 32

// ---------------------------------------------------------------------------
// Zero-fill (d_out / scatter targets are poisoned by the harness)
// ---------------------------------------------------------------------------
__global__ void zero_f32(float* __restrict__ p, int n) {
    int i = blockIdx.x * blockDim.x + threadIdx.x;
    if (i < n) p[i] = 0.0f;
}

// ---------------------------------------------------------------------------
// K1: h0 = dropout(x) @ W0 + b0   using V_WMMA_F32_16X16X4_F32 (full fp32)
// 8 waves / block, each wave computes one 16x64 row-tile.
// A frag (16x4 f32): lanes 0-15 -> K=k0,k0+1 ; lanes 16-31 -> K=k0+2,k0+3
// C/D frag (16x16 f32): vgpr v, half h: element [v+8h][lane&15]
// ---------------------------------------------------------------------------
__global__ __launch_bounds__(256)
void gemm0_dropout_wmma(const float* __restrict__ x, const float* __restrict__ u,
                        const float* __restrict__ W0, const float* __restrict__ b0,
                        float* __restrict__ h0) {
    __shared__ __align__(16) float sW[IN_DIM * HD_DIM];   // 32 KB of 320 KB/WGP
    const int tid = threadIdx.x;
    for (int i = tid * 4; i < IN_DIM * HD_DIM; i += 256 * 4)
        *(float4*)&sW[i] = *(const float4*)&W0[i];
    __syncthreads();

    const int wave = tid >> 5;
    const int lane = tid & 31;
    const int half = lane >> 4;
    const int lr   = lane & 15;
    const int row0 = (blockIdx.x * 8 + wave) * 16;        // 6250 tiles total
    if (row0 >= N_NODES) return;

    const float* __restrict__ xr = x + (size_t)(row0 + lr) * IN_DIM;
    const float* __restrict__ ur = u + (size_t)(row0 + lr) * IN_DIM;

    v8f acc[4] = {};                                      // 16x64 tile
    for (int k0 = 0; k0 < IN_DIM; k0 += 4) {
        const int ka = k0 + 2 * half;
        const float x0 = xr[ka], x1 = xr[ka + 1];
        const float u0 = ur[ka], u1 = ur[ka + 1];
        v2f a;
        a.x = (u0 >= 0.5f) ? 2.0f * x0 : 0.0f;            // dropout p=0.5, scale 2
        a.y = (u1 >= 0.5f) ? 2.0f * x1 : 0.0f;
#pragma unroll
        for (int nt = 0; nt < 4; ++nt) {
            v2f b;
            b.x = sW[ka * HD_DIM + nt * 16 + lr];
            b.y = sW[(ka + 1) * HD_DIM + nt * 16 + lr];
            acc[nt] = __builtin_amdgcn_wmma_f32_16x16x4_f32(
                false, a, false, b, (short)0, acc[nt], false, false);
        }
    }
#pragma unroll
    for (int nt = 0; nt < 4; ++nt) {
        const int n = nt * 16 + lr;
        const float bias = b0[n];
#pragma unroll
        for (int v = 0; v < 8; ++v)
            h0[(size_t)(row0 + v + 8 * half) * HD_DIM + n] = acc[nt][v] + bias;
    }
}

// ---------------------------------------------------------------------------
// K3: h2 = relu(h1) @ W1 + b1    (same WMMA scheme, K=64, N=32)
// ---------------------------------------------------------------------------
__global__ __launch_bounds__(256)
void gemm1_relu_wmma(const float* __restrict__ h1, const float* __restrict__ W1,
                     const float* __restrict__ b1, float* __restrict__ h2) {
    __shared__ __align__(16) float sW[HD_DIM * OUT_DIM];  // 8 KB
    const int tid = threadIdx.x;
    for (int i = tid * 4; i < HD_DIM * OUT_DIM; i += 256 * 4)
        *(float4*)&sW[i] = *(const float4*)&W1[i];
    __syncthreads();

    const int wave = tid >> 5;
    const int lane = tid & 31;
    const int half = lane >> 4;
    const int lr   = lane & 15;
    const int row0 = (blockIdx.x * 8 + wave) * 16;
    if (row0 >= N_NODES) return;

    const float* __restrict__ hr = h1 + (size_t)(row0 + lr) * HD_DIM;

    v8f acc[2] = {};
    for (int k0 = 0; k0 < HD_DIM; k0 += 4) {
        const int ka = k0 + 2 * half;
        v2f a;
        a.x = fmaxf(hr[ka], 0.0f);                        // fused relu
        a.y = fmaxf(hr[ka + 1], 0.0f);
#pragma unroll
        for (int nt = 0; nt < 2; ++nt) {
            v2f b;
            b.x = sW[ka * OUT_DIM + nt * 16 + lr];
            b.y = sW[(ka + 1) * OUT_DIM + nt * 16 + lr];
            acc[nt] = __builtin_amdgcn_wmma_f32_16x16x4_f32(
                false, a, false, b, (short)0, acc[nt], false, false);
        }
    }
#pragma unroll
    for (int nt = 0; nt < 2; ++nt) {
        const int n = nt * 16 + lr;
        const float bias = b1[n];
#pragma unroll
        for (int v = 0; v < 8; ++v)
            h2[(size_t)(row0 + v + 8 * half) * OUT_DIM + n] = acc[nt][v] + bias;
    }
}

// ---------------------------------------------------------------------------
// K2/K4: SpMM scatter, one wave32 per edge: out[row] += val * h[col]
// lanes cover consecutive dims -> coalesced gathers + coalesced f32 atomics,
// all hitting the 192 MB L2 (feature matrices are 12.8-25.6 MB).
// ---------------------------------------------------------------------------
template <int D>
__global__ __launch_bounds__(256)
void spmm_scatter(const float* __restrict__ val, const int* __restrict__ rowi,
                  const int* __restrict__ coli, const float* __restrict__ h,
                  float* __restrict__ out) {
    const int w    = (blockIdx.x * 256 + threadIdx.x) >> 5;   // edge id
    const int lane = threadIdx.x & 31;
    if (w >= N_EDGES) return;
    const float v = val[w];                                   // wave-uniform -> scalar
    const int   r = rowi[w];
    const int   c = coli[w];
    const float* __restrict__ src = h + (size_t)c * D;
    float* __restrict__       dst = out + (size_t)r * D;
#pragma unroll
    for (int d = lane; d < D; d += 32)
        atomicAdd(&dst[d], v * src[d]);
}

// ---------------------------------------------------------------------------
extern "C" void kernel_launch(void* const* d_in, const int* in_sizes, int n_in,
                              void* d_out, int out_size, void* d_ws, size_t ws_size,
                              hipStream_t stream) {
    const float* x      = (const float*)d_in[0];
    const float* drop_u = (const float*)d_in[1];
    const float* adj_v  = (const float*)d_in[2];
    const int*   adj_r  = (const int*)d_in[3];
    const int*   adj_c  = (const int*)d_in[4];
    const float* W0     = (const float*)d_in[5];
    const float* b0     = (const float*)d_in[6];
    const float* W1     = (const float*)d_in[7];
    const float* b1     = (const float*)d_in[8];
    float*       out    = (float*)d_out;

    float* h0 = (float*)d_ws;                         // N x 64
    float* h1 = h0 + (size_t)N_NODES * HD_DIM;        // N x 64
    float* h2 = h1 + (size_t)N_NODES * HD_DIM;        // N x 32   (64 MB total)

    const int gemmBlocks = (N_NODES / 16 + 7) / 8;    // 6250 tiles, 8 waves/block
    const int edgeBlocks = (N_EDGES * 32 + 255) / 256;

    // stage 1: dropout + GEMM0 (+ zero SpMM1 target in parallel)
    zero_f32<<<(N_NODES * HD_DIM + 255) / 256, 256, 0, stream>>>(h1, N_NODES * HD_DIM);
    gemm0_dropout_wmma<<<gemmBlocks, 256, 0, stream>>>(x, drop_u, W0, b0, h0);
    // stage 2: SpMM1 scatter
    spmm_scatter<HD_DIM><<<edgeBlocks, 256, 0, stream>>>(adj_v, adj_r, adj_c, h0, h1);
    // stage 3: relu + GEMM1 (+ zero final output)
    zero_f32<<<(N_NODES * OUT_DIM + 255) / 256, 256, 0, stream>>>(out, N_NODES * OUT_DIM);
    gemm1_relu_wmma<<<gemmBlocks, 256, 0, stream>>>(h1, W1, b1, h2);
    // stage 4: SpMM2 scatter into output
    spmm_scatter<OUT_DIM><<<edgeBlocks, 256, 0, stream>>>(adj_v, adj_r, adj_c, h2, out);
}